// BahdanauAttention_360777253257
// MI455X (gfx1250) — compile-verified
//
#include <hip/hip_runtime.h>
#include <hip/hip_bf16.h>
#include <float.h>

// Problem dims (compile-time, from reference)
#define B_ 4
#define T_ 256
#define S_ 1024
#define H_ 256

typedef __attribute__((ext_vector_type(2))) float v2f;
typedef __attribute__((ext_vector_type(4))) float v4f;
typedef __attribute__((ext_vector_type(8))) float v8f;

#define GEMM_WAVES 4   // waves per block; each wave owns a 16x32 C strip

// Native gfx1250 transcendental tanh. The compiler can't see hazard info for
// inline asm, so include the ISA-required single independent op (v_nop)
// before the result may be consumed (CDNA5 ISA trans-op co-execution rule).
__device__ __forceinline__ float htanh_f32(float x) {
    float y;
    asm volatile("v_tanh_f32 %0, %1\n\tv_nop" : "=v"(y) : "v"(x));
    return y;
}

// ---------------------------------------------------------------------------
// Batched f32 GEMM, C = A(MxK) * B(KxN). One 16x32 C strip per wave32 using
// V_WMMA_F32_16X16X4_F32 (two accumulators share the A fragment -> half the
// A traffic, two independent WMMA chains). M,N multiples of 16 (N of 32*GEMM_WAVES
// per grid.y step); K multiple of 4.
// grid = (M/16, N/(32*GEMM_WAVES), batches), block = 32*GEMM_WAVES.
// ---------------------------------------------------------------------------
__global__ __launch_bounds__(32 * GEMM_WAVES)
void gemm16_wmma_f32(const float* __restrict__ A, int lda, long long sA,
                     const float* __restrict__ Bm, int ldb, long long sB,
                     float* __restrict__ C, int ldc, long long sC,
                     int K)
{
    const int lane = threadIdx.x & 31;
    const int wid  = threadIdx.x >> 5;
    const int m0   = blockIdx.x * 16;
    const int n0   = (blockIdx.y * GEMM_WAVES + wid) * 32;
    const long long bb = blockIdx.z;

    A  += bb * sA;
    Bm += bb * sB;
    C  += bb * sC;

    // A fragment (16x4 f32): lanes 0-15 -> rows m0+lane, K pair {k,k+1};
    //                        lanes 16-31 -> rows m0+lane-16, K pair {k+2,k+3}.
    const int ar   = m0 + (lane & 15);
    const int kgrp = (lane >> 4) << 1;          // 0 or 2
    // B fragment (4x16 f32): column n + (lane&15), same K pair split.
    const int bn   = n0 + (lane & 15);

    v8f acc0 = {}, acc1 = {};
    for (int k = 0; k < K; k += 4) {
        v2f a = *(const v2f*)(A + (long long)ar * lda + (k + kgrp));
        const float* brow0 = Bm + (long long)(k + kgrp) * ldb;
        const float* brow1 = brow0 + ldb;
        v2f b0, b1;
        b0.x = brow0[bn];      b0.y = brow1[bn];
        b1.x = brow0[bn + 16]; b1.y = brow1[bn + 16];
        acc0 = __builtin_amdgcn_wmma_f32_16x16x4_f32(
            false, a, false, b0, (short)0, acc0, false, false);
        acc1 = __builtin_amdgcn_wmma_f32_16x16x4_f32(
            false, a, false, b1, (short)0, acc1, false, false);
    }

    // C/D layout: VGPR i holds row m0 + i (+8 for lanes 16-31), col n+(lane&15)
    const int rbase = m0 + ((lane >> 4) << 3);
    #pragma unroll
    for (int i = 0; i < 8; ++i) {
        C[(long long)(rbase + i) * ldc + bn]      = acc0[i];
        C[(long long)(rbase + i) * ldc + bn + 16] = acc1[i];
    }
}

// ---------------------------------------------------------------------------
// Fused additive score + mask + softmax.
// One block (256 threads = 8 waves) per (b, t). Each wave owns one s at a
// time; lanes cover 8 contiguous h each (coalesced 32B/lane loads), native
// v_tanh_f32 for the additive feature, shfl_xor wave reduction, then
// block-wide masked softmax in LDS. Writes attention weights (B,T,S) to ws.
// ---------------------------------------------------------------------------
__global__ __launch_bounds__(256)
void scores_softmax(const float* __restrict__ qp, const float* __restrict__ ep,
                    const float* __restrict__ vvec, const int* __restrict__ srclen,
                    float* __restrict__ attn)
{
    __shared__ float qs[H_];
    __shared__ float vs[H_];
    __shared__ float sc[S_];
    __shared__ float red[256];

    const int t    = blockIdx.x;
    const int b    = blockIdx.y;
    const int tid  = threadIdx.x;
    const int lane = tid & 31;
    const int wid  = tid >> 5;

    qs[tid] = qp[((long long)(b * T_ + t)) * H_ + tid];
    vs[tid] = vvec[tid];
    __syncthreads();

    // Hoist this lane's q-projection and v slices into registers.
    const int hb = lane * 8;
    float qv[8], vv[8];
    #pragma unroll
    for (int j = 0; j < 8; ++j) { qv[j] = qs[hb + j]; vv[j] = vs[hb + j]; }

    const float* eb = ep + (long long)b * S_ * H_;

    for (int s = wid; s < S_; s += 8) {
        const float* row = eb + (long long)s * H_ + hb;
        if (s + 8 < S_) __builtin_prefetch(row + 8 * H_, 0, 1);  // global_prefetch_b8
        v4f e0 = *(const v4f*)(row);
        v4f e1 = *(const v4f*)(row + 4);
        float ev[8] = {e0.x, e0.y, e0.z, e0.w, e1.x, e1.y, e1.z, e1.w};
        float th[8];
        #pragma unroll
        for (int j = 0; j < 8; ++j)
            th[j] = htanh_f32(qv[j] + ev[j]);   // v_tanh_f32
        float acc = 0.f;
        #pragma unroll
        for (int j = 0; j < 8; ++j)
            acc = fmaf(vv[j], th[j], acc);
        // wave32 reduction
        #pragma unroll
        for (int off = 16; off > 0; off >>= 1)
            acc += __shfl_xor(acc, off, 32);
        if (lane == 0) sc[s] = acc;
    }
    __syncthreads();

    // Mask + block max
    const int L = srclen[b];
    float lmax = -FLT_MAX;
    #pragma unroll
    for (int s = tid; s < S_; s += 256) {
        float x = (s < L) ? sc[s] : -FLT_MAX;
        sc[s] = x;
        lmax = fmaxf(lmax, x);
    }
    red[tid] = lmax;
    __syncthreads();
    for (int off = 128; off > 0; off >>= 1) {
        if (tid < off) red[tid] = fmaxf(red[tid], red[tid + off]);
        __syncthreads();
    }
    const float mx = red[0];
    __syncthreads();

    // exp + block sum
    float lsum = 0.f;
    #pragma unroll
    for (int s = tid; s < S_; s += 256) {
        float e = __expf(sc[s] - mx);   // masked positions -> exp(-inf)=0
        sc[s] = e;
        lsum += e;
    }
    red[tid] = lsum;
    __syncthreads();
    for (int off = 128; off > 0; off >>= 1) {
        if (tid < off) red[tid] += red[tid + off];
        __syncthreads();
    }
    const float inv = 1.0f / red[0];

    float* arow = attn + ((long long)(b * T_ + t)) * S_;
    #pragma unroll
    for (int s = tid; s < S_; s += 256)
        arow[s] = sc[s] * inv;
}

extern "C" void kernel_launch(void* const* d_in, const int* in_sizes, int n_in,
                              void* d_out, int out_size, void* d_ws, size_t ws_size,
                              hipStream_t stream) {
    const float* query = (const float*)d_in[0];   // (B,T,H)
    const float* enc   = (const float*)d_in[1];   // (B,S,H)
    const int*   slen  = (const int*)d_in[2];     // (B,)
    const float* W_h   = (const float*)d_in[3];   // (H,H)
    const float* W_s   = (const float*)d_in[4];   // (H,H)
    const float* vvec  = (const float*)d_in[5];   // (H,)
    float* out = (float*)d_out;                   // (B,T,H)

    float* qp   = (float*)d_ws;                         // B*T*H
    float* epj  = qp  + (size_t)B_ * T_ * H_;           // B*S*H
    float* attn = epj + (size_t)B_ * S_ * H_;           // B*T*S

    dim3 gblk(32 * GEMM_WAVES);
    const int ncols = 32 * GEMM_WAVES;                  // columns per block

    // 1) q_proj = query @ W_h   (flat M = B*T = 1024, N = 256, K = 256)
    gemm16_wmma_f32<<<dim3((B_ * T_) / 16, H_ / ncols, 1), gblk, 0, stream>>>(
        query, H_, 0, W_h, H_, 0, qp, H_, 0, H_);

    // 2) e_proj = enc @ W_s     (flat M = B*S = 4096, N = 256, K = 256)
    gemm16_wmma_f32<<<dim3((B_ * S_) / 16, H_ / ncols, 1), gblk, 0, stream>>>(
        enc, H_, 0, W_s, H_, 0, epj, H_, 0, H_);

    // 3) fused score + mask + softmax -> attn (B,T,S)
    scores_softmax<<<dim3(T_, B_), 256, 0, stream>>>(qp, epj, vvec, slen, attn);

    // 4) attn_out = attn @ enc  (per-batch: M=T=256, N=H=256, K=S=1024)
    gemm16_wmma_f32<<<dim3(T_ / 16, H_ / ncols, B_), gblk, 0, stream>>>(
        attn, S_, (long long)T_ * S_,
        enc,  H_, (long long)S_ * H_,
        out,  H_, (long long)T_ * H_,
        S_);
}